// WildcatPool2d_42812234006995
// MI455X (gfx1250) — compile-verified
//
#include <hip/hip_runtime.h>
#include <stdint.h>

#define N_ELEM 1024      // 32*32 per (b,c) row
#define KSEL   205       // round(0.2 * 1024)
#define ALPHA_F 0.7f

// Async global->LDS path (gfx1250): guard so compile never breaks if the
// builtin is absent on this toolchain.
#if defined(__HIP_DEVICE_COMPILE__) && \
    __has_builtin(__builtin_amdgcn_global_load_async_to_lds_b128) && \
    __has_builtin(__builtin_amdgcn_s_wait_asynccnt)
#define WCP_ASYNC 1
#else
#define WCP_ASYNC 0
#endif

#if WCP_ASYNC
typedef int wcp_v4i __attribute__((ext_vector_type(4)));
typedef __attribute__((address_space(1))) wcp_v4i* wcp_gptr;
typedef __attribute__((address_space(3))) wcp_v4i* wcp_lptr;
#endif

// Order-preserving float -> uint32 key (monotonic increasing).
__device__ __forceinline__ uint32_t f2k(float f) {
    uint32_t u = __float_as_uint(f);
    return (u & 0x80000000u) ? ~u : (u | 0x80000000u);
}
__device__ __forceinline__ float k2f(uint32_t k) {
    return __uint_as_float((k & 0x80000000u) ? (k & 0x7FFFFFFFu) : ~k);
}

// Exact radix-select (descending) of the k-th largest transformed key among
// the 1024 elements held by this block (4 per thread). invmask==0 selects on
// the value order; invmask==0xFFFFFFFF selects on reversed order (bottom-k).
// Outputs: tkey (transformed key of k-th largest), c_gt / s_gt = count & sum
// of elements whose transformed key is strictly greater than tkey.
__device__ __forceinline__ void radix_select_desc(
    const uint32_t key[4], const float val[4], uint32_t invmask, uint32_t k,
    uint32_t* s_cnt, float* s_sum,
    uint32_t* s_bb, uint32_t* s_cab, float* s_sab,
    uint32_t& out_tkey, uint32_t& out_cgt, float& out_sgt)
{
    const int tid = threadIdx.x;
    uint32_t prefix = 0, kk = k, cgt = 0;
    float sgt = 0.0f;

    for (int pass = 0; pass < 4; ++pass) {
        const int shift = 24 - 8 * pass;
        s_cnt[tid] = 0u;
        s_sum[tid] = 0.0f;
        __syncthreads();

        #pragma unroll
        for (int e = 0; e < 4; ++e) {
            uint32_t tk = key[e] ^ invmask;
            uint32_t hi = (pass == 0) ? 0u : (tk >> (shift + 8));
            if (hi == prefix) {
                uint32_t b = (tk >> shift) & 0xFFu;
                atomicAdd(&s_cnt[b], 1u);
                atomicAdd(&s_sum[b], val[e]);
            }
        }
        __syncthreads();

        // Wave 0: descending suffix scan over 256 bins to find the bucket
        // where the cumulative count crosses kk.
        if (tid < 32) {
            const int lane = tid;
            uint32_t cl[8]; float sl[8];
            uint32_t csum = 0; float fsum = 0.0f;
            #pragma unroll
            for (int j = 0; j < 8; ++j) {
                int b = 255 - (lane * 8 + j);       // descending order
                cl[j] = s_cnt[b];
                sl[j] = s_sum[b];
                csum += cl[j];
                fsum += sl[j];
            }
            // inclusive wave scan of per-lane partials
            uint32_t cinc = csum; float finc = fsum;
            #pragma unroll
            for (int d = 1; d < 32; d <<= 1) {
                uint32_t cp = __shfl_up(cinc, d, 32);
                float    fp = __shfl_up(finc, d, 32);
                if (lane >= d) { cinc += cp; finc += fp; }
            }
            uint32_t cum = cinc - csum;   // exclusive (count above this lane's bins)
            float   fcum = finc - fsum;
            #pragma unroll
            for (int j = 0; j < 8; ++j) {
                uint32_t c = cl[j];
                if (cum < kk && cum + c >= kk) {   // unique crossing bucket
                    *s_bb  = (uint32_t)(255 - (lane * 8 + j));
                    *s_cab = cum;    // strictly above the chosen bucket
                    *s_sab = fcum;
                }
                cum  += c;
                fcum += sl[j];
            }
        }
        __syncthreads();

        uint32_t b   = *s_bb;
        uint32_t cab = *s_cab;
        float    sab = *s_sab;
        cgt += cab;
        sgt += sab;
        kk  -= cab;
        prefix = (prefix << 8) | b;
        __syncthreads();   // protect s_cnt/s_sum/broadcast before next pass zeroing
    }
    out_tkey = prefix;
    out_cgt  = cgt;
    out_sgt  = sgt;
}

__global__ __launch_bounds__(256)
void wildcat_pool2d_kernel(const float* __restrict__ x, float* __restrict__ out)
{
    __shared__ uint32_t s_cnt[256];
    __shared__ float    s_sum[256];
    __shared__ float    s_stage[N_ELEM];
    __shared__ uint32_t s_bb;
    __shared__ uint32_t s_cab;
    __shared__ float    s_sab;

    const int tid = threadIdx.x;
    const long long row = blockIdx.x;
    const float* base = x + row * (long long)N_ELEM;

    float v[4];
#if WCP_ASYNC
    {
        const float* gp = base + tid * 4;
        float* lp = &s_stage[tid * 4];
        __builtin_amdgcn_global_load_async_to_lds_b128(
            (wcp_gptr)gp,
            (wcp_lptr)lp,
            0, 0);
        __builtin_amdgcn_s_wait_asynccnt(0);
        __syncthreads();
        v[0] = s_stage[tid * 4 + 0];
        v[1] = s_stage[tid * 4 + 1];
        v[2] = s_stage[tid * 4 + 2];
        v[3] = s_stage[tid * 4 + 3];
        __syncthreads();   // done with s_stage before histograms reuse LDS
    }
#else
    {
        (void)s_stage;
        const float4 vv = *reinterpret_cast<const float4*>(base + tid * 4);
        v[0] = vv.x; v[1] = vv.y; v[2] = vv.z; v[3] = vv.w;
    }
#endif

    uint32_t key[4];
    #pragma unroll
    for (int e = 0; e < 4; ++e) key[e] = f2k(v[e]);

    // ---- top-KSEL ----
    uint32_t tk_hi, c_hi; float s_hi;
    radix_select_desc(key, v, 0u, KSEL,
                      s_cnt, s_sum, &s_bb, &s_cab, &s_sab,
                      tk_hi, c_hi, s_hi);
    float T_hi = k2f(tk_hi);
    float sum_top = s_hi + (float)(KSEL - c_hi) * T_hi;   // top_k tie semantics

    // ---- bottom-KSEL (same select on inverted keys) ----
    uint32_t tk_lo, c_lo; float s_lo;
    radix_select_desc(key, v, 0xFFFFFFFFu, KSEL,
                      s_cnt, s_sum, &s_bb, &s_cab, &s_sab,
                      tk_lo, c_lo, s_lo);
    float T_lo = k2f(tk_lo ^ 0xFFFFFFFFu);                // undo inversion
    float sum_bot = s_lo + (float)(KSEL - c_lo) * T_lo;

    if (tid == 0) {
        float top_avg  = sum_top * (1.0f / (float)KSEL);
        float bot_term = sum_bot * (ALPHA_F / (float)KSEL);
        out[row] = 0.5f * (top_avg + bot_term);
    }
}

extern "C" void kernel_launch(void* const* d_in, const int* in_sizes, int n_in,
                              void* d_out, int out_size, void* d_ws, size_t ws_size,
                              hipStream_t stream)
{
    (void)n_in; (void)d_ws; (void)ws_size;
    const float* x = (const float*)d_in[0];
    float* out = (float*)d_out;
    const int rows = in_sizes[0] / N_ELEM;   // 64*512 = 32768; matches out_size
    (void)out_size;
    wildcat_pool2d_kernel<<<dim3(rows), dim3(256), 0, stream>>>(x, out);
}